// SimpleGAT_88888643158268
// MI455X (gfx1250) — compile-verified
//
#include <hip/hip_runtime.h>
#include <math.h>

typedef __attribute__((ext_vector_type(2))) float v2f;
typedef __attribute__((ext_vector_type(8))) float v8f;

#define N_NODES 50000
#define N_EDGES 400000
#define TOT_E   (2 * N_EDGES + N_NODES)   /* 850000 */
#define HID     256
#define HEADS   8
#define DPH     32
#define OUT_DIM 128
#define SEG_EPS 1e-12f
#define BN_EPS  1e-5f

// ---------------------------------------------------------------- utilities

__global__ void zero_kernel(float* __restrict__ p, long n) {
  long i = (long)blockIdx.x * blockDim.x + threadIdx.x;
  long stride = (long)gridDim.x * blockDim.x;
  for (; i < n; i += stride) p[i] = 0.0f;
}

// Map a flattened edge id e in [0, TOT_E) onto (src, dst) of the
// symmetrized + self-loop edge list without materializing it.
__device__ __forceinline__ void edge_endpoints(const int* __restrict__ ei,
                                               int e, int& src, int& dst) {
  if (e < N_EDGES) {                      // forward edges
    src = ei[e];
    dst = ei[N_EDGES + e];
  } else if (e < 2 * N_EDGES) {           // reversed edges
    int t = e - N_EDGES;
    src = ei[N_EDGES + t];
    dst = ei[t];
  } else {                                // self loops
    int t = e - 2 * N_EDGES;
    src = t;
    dst = t;
  }
}

// ------------------------------------------------- fp32 WMMA GEMM (+ bias)
// C[M x NC] = A[M x K] @ B[K x NC] + bias, all fp32, K/NC compile-time so
// every inner-loop load uses an immediate offset (no per-iter address math).
//
// Block = (NC/32) waves; wave w computes TWO adjacent 16x16 tiles at
// columns w*32 and w*32+16 of node-row-tile blockIdx.x*16, so each A
// fragment feeds two V_WMMA_F32_16X16X4_F32 ops.
//
// Fragment layouts (CDNA5 ISA 7.12.2):
//   A 16x4 : lane m=lane&15, VGPR j holds K = (lane>>4)*2 + j
//   B 4x16 : lane n=lane&15, VGPR j holds K = (lane>>4)*2 + j
//   C 16x16: VGPR i -> row m0 + i + 8*(lane>>4), col n0 + (lane&15)
template <int K, int NC>
__global__ void gemm_wmma_kernel(const float* __restrict__ A,
                                 const float* __restrict__ B,
                                 const float* __restrict__ bias,
                                 float* __restrict__ C) {
  const int wave = threadIdx.x >> 5;
  const int lane = threadIdx.x & 31;
  const int m0 = blockIdx.x << 4;
  const int n0 = wave << 5;               // 32 columns per wave

  const int rowA  = m0 + (lane & 15);
  const int kh    = (lane >> 4) << 1;     // 0 or 2
  const int colB0 = n0 + (lane & 15);
  const int colB1 = colB0 + 16;

  const float bv0 = bias[colB0];
  const float bv1 = bias[colB1];
  v8f acc0, acc1;
#pragma unroll
  for (int i = 0; i < 8; ++i) { acc0[i] = bv0; acc1[i] = bv1; }

  const float* ap = A + (long)rowA * K + kh;
  const float* bp = B + (long)kh * NC;

#pragma unroll 8
  for (int k = 0; k < K; k += 4) {
    v2f a, b0, b1;
    a.x  = ap[k];
    a.y  = ap[k + 1];
    b0.x = bp[k * NC + colB0];
    b0.y = bp[(k + 1) * NC + colB0];
    b1.x = bp[k * NC + colB1];
    b1.y = bp[(k + 1) * NC + colB1];
    acc0 = __builtin_amdgcn_wmma_f32_16x16x4_f32(
        false, a, false, b0, (short)0, acc0, false, false);
    acc1 = __builtin_amdgcn_wmma_f32_16x16x4_f32(
        false, a, false, b1, (short)0, acc1, false, false);
  }

  const int rowC = m0 + ((lane >> 4) << 3);
#pragma unroll
  for (int i = 0; i < 8; ++i) {
    C[(long)(rowC + i) * NC + colB0] = acc0[i];
    C[(long)(rowC + i) * NC + colB1] = acc1[i];
  }
}

// ------------------------------------------ per-node attention score halves
// s_l[n][h] = sum_d H[n][h*32+d] * a_l[h][d]   (and same for s_r)
// One wave per node; lane d = channel within head; wave32 shfl_xor reduce.
__global__ void scores_kernel(const float* __restrict__ H,
                              const float* __restrict__ a_l,
                              const float* __restrict__ a_r,
                              float* __restrict__ s_l,
                              float* __restrict__ s_r) {
  const int wave = threadIdx.x >> 5;
  const int lane = threadIdx.x & 31;
  const int node = blockIdx.x * (blockDim.x >> 5) + wave;
  if (node >= N_NODES) return;
  const float* hrow = H + (long)node * HID;
#pragma unroll
  for (int h = 0; h < HEADS; ++h) {
    const float hv = hrow[h * DPH + lane];
    float vl = hv * a_l[h * DPH + lane];
    float vr = hv * a_r[h * DPH + lane];
#pragma unroll
    for (int off = 16; off > 0; off >>= 1) {
      vl += __shfl_xor(vl, off, 32);
      vr += __shfl_xor(vr, off, 32);
    }
    if (lane == 0) {
      s_l[node * HEADS + h] = vl;
      s_r[node * HEADS + h] = vr;
    }
  }
}

// --------------------------------- edge pass 1: exp(logit) + denom scatter
__global__ void edge_exp_kernel(const int* __restrict__ ei,
                                const float* __restrict__ s_l,
                                const float* __restrict__ s_r,
                                float* __restrict__ e_buf,
                                float* __restrict__ denom) {
  const int t = blockIdx.x * blockDim.x + threadIdx.x;
  if (t >= TOT_E * HEADS) return;
  const int e = t >> 3;
  const int h = t & 7;
  int src, dst;
  edge_endpoints(ei, e, src, dst);
  // reference uses unstabilized exp
  const float ex = expf(s_l[src * HEADS + h] + s_r[dst * HEADS + h]);
  e_buf[t] = ex;
  atomicAdd(&denom[dst * HEADS + h], ex);
}

// ------------------------- edge pass 2: alpha-weighted message scatter-add
// Block = 256 threads = one edge slice; thread = (head = tid>>5, chan =
// tid&31); grid-stride over edges (8 per block) to amortize dispatch.
__global__ void edge_scatter_kernel(const int* __restrict__ ei,
                                    const float* __restrict__ H,
                                    const float* __restrict__ e_buf,
                                    const float* __restrict__ denom,
                                    float* __restrict__ out) {
  const int h = threadIdx.x >> 5;
  const int c = threadIdx.x & 31;
  for (int e = blockIdx.x; e < TOT_E; e += gridDim.x) {
    int src, dst;
    edge_endpoints(ei, e, src, dst);
    const float alpha =
        e_buf[e * HEADS + h] / (denom[dst * HEADS + h] + SEG_EPS);
    const float msg = H[(long)src * HID + h * DPH + c] * alpha;
    atomicAdd(&out[(long)dst * HID + h * DPH + c], msg);
  }
}

// --------------------------------------------------- BatchNorm (training)
// Thread = column (256 threads), block sweeps a row range: fully coalesced.
__global__ void bn_stats_kernel(const float* __restrict__ X,
                                float* __restrict__ sums,
                                float* __restrict__ sumsq) {
  const int c = threadIdx.x;
  const int rows_per_block = (N_NODES + gridDim.x - 1) / gridDim.x;
  const int r0 = blockIdx.x * rows_per_block;
  int r1 = r0 + rows_per_block;
  if (r1 > N_NODES) r1 = N_NODES;
  float s = 0.0f, sq = 0.0f;
  for (int r = r0; r < r1; ++r) {
    const float v = X[(long)r * HID + c];
    s += v;
    sq += v * v;
  }
  atomicAdd(&sums[c], s);
  atomicAdd(&sumsq[c], sq);
}

__global__ void bn_apply_relu_kernel(const float* __restrict__ X,
                                     const float* __restrict__ sums,
                                     const float* __restrict__ sumsq,
                                     const float* __restrict__ gamma,
                                     const float* __restrict__ beta,
                                     float* __restrict__ Y) {
  const long idx = (long)blockIdx.x * blockDim.x + threadIdx.x;
  if (idx >= (long)N_NODES * HID) return;
  const int c = (int)(idx & (HID - 1));
  const float inv_n = 1.0f / (float)N_NODES;
  const float mean = sums[c] * inv_n;
  const float var = sumsq[c] * inv_n - mean * mean;   // biased variance
  const float v = (X[idx] - mean) * rsqrtf(var + BN_EPS) * gamma[c] + beta[c];
  Y[idx] = v > 0.0f ? v : 0.0f;
}

// ------------------------------------------------------------------ driver

extern "C" void kernel_launch(void* const* d_in, const int* in_sizes, int n_in,
                              void* d_out, int out_size, void* d_ws, size_t ws_size,
                              hipStream_t stream) {
  const float* x   = (const float*)d_in[0];
  const int*   ei  = (const int*)d_in[1];
  const float* W1  = (const float*)d_in[2];
  const float* b1  = (const float*)d_in[3];
  const float* al1 = (const float*)d_in[4];
  const float* ar1 = (const float*)d_in[5];
  const float* g1  = (const float*)d_in[6];
  const float* be1 = (const float*)d_in[7];
  const float* W2  = (const float*)d_in[8];
  const float* b2  = (const float*)d_in[9];
  const float* al2 = (const float*)d_in[10];
  const float* ar2 = (const float*)d_in[11];
  const float* g2  = (const float*)d_in[12];
  const float* be2 = (const float*)d_in[13];
  const float* Wh  = (const float*)d_in[14];
  const float* bh  = (const float*)d_in[15];
  float* out = (float*)d_out;

  // workspace carve-up (all fp32)
  const long FEAT = (long)N_NODES * HID;               // 12.8M floats
  char* ws = (char*)d_ws;
  float* bufA  = (float*)(ws);                         // 51.2 MB
  float* bufB  = (float*)(ws + FEAT * sizeof(float));  // 51.2 MB
  float* s_l   = bufB + FEAT;
  float* s_r   = s_l + (long)N_NODES * HEADS;
  float* denom = s_r + (long)N_NODES * HEADS;
  float* e_buf = denom + (long)N_NODES * HEADS;        // 27.2 MB
  float* sums  = e_buf + (long)TOT_E * HEADS;
  float* sumsq = sums + HID;

  const int gemm_grid = N_NODES / 16;                  // 3125 row tiles
  const int score_blocks = (N_NODES + 7) / 8;
  const int exp_blocks = (TOT_E * HEADS + 255) / 256;
  const int scatter_blocks = TOT_E / 8;                // 8 edges per block
  const int bn_apply_blocks = (int)((FEAT + 255) / 256);

  // ---------------- layer 1 ----------------
  gemm_wmma_kernel<HID, HID><<<gemm_grid, 256, 0, stream>>>(x, W1, b1, bufA);
  scores_kernel<<<score_blocks, 256, 0, stream>>>(bufA, al1, ar1, s_l, s_r);
  zero_kernel<<<1024, 256, 0, stream>>>(denom, (long)N_NODES * HEADS);
  zero_kernel<<<4096, 256, 0, stream>>>(bufB, FEAT);
  edge_exp_kernel<<<exp_blocks, 256, 0, stream>>>(ei, s_l, s_r, e_buf, denom);
  edge_scatter_kernel<<<scatter_blocks, 256, 0, stream>>>(ei, bufA, e_buf,
                                                          denom, bufB);
  zero_kernel<<<2, 256, 0, stream>>>(sums, 2 * HID);
  bn_stats_kernel<<<200, HID, 0, stream>>>(bufB, sums, sumsq);
  bn_apply_relu_kernel<<<bn_apply_blocks, 256, 0, stream>>>(bufB, sums, sumsq,
                                                            g1, be1, bufA);

  // ---------------- layer 2 ----------------
  gemm_wmma_kernel<HID, HID><<<gemm_grid, 256, 0, stream>>>(bufA, W2, b2, bufB);
  scores_kernel<<<score_blocks, 256, 0, stream>>>(bufB, al2, ar2, s_l, s_r);
  zero_kernel<<<1024, 256, 0, stream>>>(denom, (long)N_NODES * HEADS);
  zero_kernel<<<4096, 256, 0, stream>>>(bufA, FEAT);
  edge_exp_kernel<<<exp_blocks, 256, 0, stream>>>(ei, s_l, s_r, e_buf, denom);
  edge_scatter_kernel<<<scatter_blocks, 256, 0, stream>>>(ei, bufB, e_buf,
                                                          denom, bufA);
  zero_kernel<<<2, 256, 0, stream>>>(sums, 2 * HID);
  bn_stats_kernel<<<200, HID, 0, stream>>>(bufA, sums, sumsq);
  bn_apply_relu_kernel<<<bn_apply_blocks, 256, 0, stream>>>(bufA, sums, sumsq,
                                                            g2, be2, bufB);

  // ---------------- classification head (128 cols -> 4 waves) ----------------
  gemm_wmma_kernel<HID, OUT_DIM><<<gemm_grid, 128, 0, stream>>>(bufB, Wh, bh,
                                                                out);
}